// AdversarialModel_26242250179145
// MI455X (gfx1250) — compile-verified
//
#include <hip/hip_runtime.h>
#include <hip/hip_bf16.h>
#include <math.h>

typedef float v2f __attribute__((ext_vector_type(2)));
typedef float v8f __attribute__((ext_vector_type(8)));

#define NWAVES 8
#define TPB 256
#define TILE_J 1024

// ---------------------------------------------------------------------------
// Pass 1: amatavg[i] = (1/N) sum_j |v1[i]-v1[j]| * w[j]   (and same for v2)
// One block = 16 rows. Row sums accumulated with V_WMMA_F32_16X16X4_F32
// (A-operand = elementwise products, B = all-ones => D row-sums).
// ---------------------------------------------------------------------------
__global__ __launch_bounds__(TPB) void row_avg_kernel(
    const float* __restrict__ v1, const float* __restrict__ v2,
    const float* __restrict__ w, float* __restrict__ aavg,
    float* __restrict__ bavg, int N) {
  __shared__ float sv1[TILE_J], sv2[TILE_J], sw[TILE_J];
  __shared__ float sredA[NWAVES][16], sredB[NWAVES][16];

  const int tid  = threadIdx.x;
  const int wave = tid >> 5;
  const int lane = tid & 31;
  const int hf   = lane >> 4;   // which K-pair this lane supplies (0: K=0,1  1: K=2,3)
  const int lm   = lane & 15;   // row within tile for A-operand
  const int rowBase = blockIdx.x * 16;

  const float v1i = v1[rowBase + lm];
  const float v2i = v2[rowBase + lm];

  v2f ones; ones.x = 1.0f; ones.y = 1.0f;
  v8f accA = {};
  v8f accB = {};

  for (int t0 = 0; t0 < N; t0 += TILE_J) {
    for (int k = tid; k < TILE_J; k += TPB) {
      sv1[k] = v1[t0 + k];
      sv2[k] = v2[t0 + k];
      sw[k]  = w[t0 + k];
    }
    __syncthreads();
    const int jb = wave * (TILE_J / NWAVES);
    const int je = jb + (TILE_J / NWAVES);
    for (int jc = jb; jc < je; jc += 4) {
      const int j0 = jc + hf * 2;
      const float w0 = sw[j0], w1 = sw[j0 + 1];
      v2f mA, mB;
      mA.x = fabsf(v1i - sv1[j0])     * w0;
      mA.y = fabsf(v1i - sv1[j0 + 1]) * w1;
      mB.x = fabsf(v2i - sv2[j0])     * w0;
      mB.y = fabsf(v2i - sv2[j0 + 1]) * w1;
      accA = __builtin_amdgcn_wmma_f32_16x16x4_f32(false, mA, false, ones,
                                                   (short)0, accA, false, false);
      accB = __builtin_amdgcn_wmma_f32_16x16x4_f32(false, mB, false, ones,
                                                   (short)0, accB, false, false);
    }
    __syncthreads();
  }

  // lanes 0 (rows 0-7) and 16 (rows 8-15) hold the row sums in components 0..7
  if (lm == 0) {
    for (int r = 0; r < 8; ++r) {
      sredA[wave][hf * 8 + r] = accA[r];
      sredB[wave][hf * 8 + r] = accB[r];
    }
  }
  __syncthreads();
  if (tid < 16) {
    float sa = 0.0f, sb = 0.0f;
    for (int wv = 0; wv < NWAVES; ++wv) { sa += sredA[wv][tid]; sb += sredB[wv][tid]; }
    const float inv = 1.0f / (float)N;
    aavg[rowBase + tid] = sa * inv;
    bavg[rowBase + tid] = sb * inv;
  }
}

// ---------------------------------------------------------------------------
// abar = mean(amatavg*w), bbar = mean(bmatavg*w)
// ---------------------------------------------------------------------------
__global__ __launch_bounds__(TPB) void mean_kernel(
    const float* __restrict__ aavg, const float* __restrict__ bavg,
    const float* __restrict__ w, float* __restrict__ scal, int N) {
  __shared__ float ra[TPB], rb[TPB];
  const int tid = threadIdx.x;
  float sa = 0.0f, sb = 0.0f;
  for (int k = tid; k < N; k += TPB) {
    const float wk = w[k];
    sa += aavg[k] * wk;
    sb += bavg[k] * wk;
  }
  ra[tid] = sa; rb[tid] = sb;
  __syncthreads();
  for (int s = TPB / 2; s > 0; s >>= 1) {
    if (tid < s) { ra[tid] += ra[tid + s]; rb[tid] += rb[tid + s]; }
    __syncthreads();
  }
  if (tid == 0) {
    scal[0] = ra[0] / (float)N;
    scal[1] = rb[0] / (float)N;
  }
}

// ---------------------------------------------------------------------------
// Pass 2: per-row centered sums.
//   A_ij = |v1i-v1j| - aavg[j] - (aavg[i]-abar)
//   B_ij = |v2i-v2j| - bavg[j] - (bavg[i]-bbar)
//   ABrow[i]=|sum A*B*w|/N, AArow[i]=sum A*A*w/N, BBrow[i]=sum B*B*w/N
// ---------------------------------------------------------------------------
__global__ __launch_bounds__(TPB) void row_disco_kernel(
    const float* __restrict__ v1, const float* __restrict__ v2,
    const float* __restrict__ w, const float* __restrict__ aavg,
    const float* __restrict__ bavg, const float* __restrict__ scal,
    float* __restrict__ ABrow, float* __restrict__ AArow,
    float* __restrict__ BBrow, int N) {
  __shared__ float sv1[TILE_J], sv2[TILE_J], sw[TILE_J];
  __shared__ float saa[TILE_J], sba[TILE_J];
  __shared__ float sredAB[NWAVES][16], sredAA[NWAVES][16], sredBB[NWAVES][16];

  const int tid  = threadIdx.x;
  const int wave = tid >> 5;
  const int lane = tid & 31;
  const int hf   = lane >> 4;
  const int lm   = lane & 15;
  const int rowBase = blockIdx.x * 16;
  const int i = rowBase + lm;

  const float abar = scal[0];
  const float bbar = scal[1];
  const float v1i = v1[i];
  const float v2i = v2[i];
  const float ci = aavg[i] - abar;
  const float di = bavg[i] - bbar;

  v2f ones; ones.x = 1.0f; ones.y = 1.0f;
  v8f accAB = {};
  v8f accAA = {};
  v8f accBB = {};

  for (int t0 = 0; t0 < N; t0 += TILE_J) {
    for (int k = tid; k < TILE_J; k += TPB) {
      sv1[k] = v1[t0 + k];
      sv2[k] = v2[t0 + k];
      sw[k]  = w[t0 + k];
      saa[k] = aavg[t0 + k];
      sba[k] = bavg[t0 + k];
    }
    __syncthreads();
    const int jb = wave * (TILE_J / NWAVES);
    const int je = jb + (TILE_J / NWAVES);
    for (int jc = jb; jc < je; jc += 4) {
      const int j0 = jc + hf * 2;
      const float w0 = sw[j0], w1 = sw[j0 + 1];
      const float Ax = fabsf(v1i - sv1[j0])     - saa[j0]     - ci;
      const float Ay = fabsf(v1i - sv1[j0 + 1]) - saa[j0 + 1] - ci;
      const float Bx = fabsf(v2i - sv2[j0])     - sba[j0]     - di;
      const float By = fabsf(v2i - sv2[j0 + 1]) - sba[j0 + 1] - di;
      v2f mAB, mAA, mBB;
      mAB.x = Ax * Bx * w0;  mAB.y = Ay * By * w1;
      mAA.x = Ax * Ax * w0;  mAA.y = Ay * Ay * w1;
      mBB.x = Bx * Bx * w0;  mBB.y = By * By * w1;
      accAB = __builtin_amdgcn_wmma_f32_16x16x4_f32(false, mAB, false, ones,
                                                    (short)0, accAB, false, false);
      accAA = __builtin_amdgcn_wmma_f32_16x16x4_f32(false, mAA, false, ones,
                                                    (short)0, accAA, false, false);
      accBB = __builtin_amdgcn_wmma_f32_16x16x4_f32(false, mBB, false, ones,
                                                    (short)0, accBB, false, false);
    }
    __syncthreads();
  }

  if (lm == 0) {
    for (int r = 0; r < 8; ++r) {
      sredAB[wave][hf * 8 + r] = accAB[r];
      sredAA[wave][hf * 8 + r] = accAA[r];
      sredBB[wave][hf * 8 + r] = accBB[r];
    }
  }
  __syncthreads();
  if (tid < 16) {
    float sab = 0.0f, saa2 = 0.0f, sbb = 0.0f;
    for (int wv = 0; wv < NWAVES; ++wv) {
      sab  += sredAB[wv][tid];
      saa2 += sredAA[wv][tid];
      sbb  += sredBB[wv][tid];
    }
    const float inv = 1.0f / (float)N;
    ABrow[rowBase + tid] = fabsf(sab) * inv;
    AArow[rowBase + tid] = saa2 * inv;
    BBrow[rowBase + tid] = sbb * inv;
  }
}

// ---------------------------------------------------------------------------
// Focal BCE reductions + final combine (single block, deterministic).
// ---------------------------------------------------------------------------
__global__ __launch_bounds__(TPB) void final_kernel(
    const float* __restrict__ target, const float* __restrict__ output,
    const float* __restrict__ y_class, const float* __restrict__ y_pred_class,
    const float* __restrict__ w, const float* __restrict__ ABrow,
    const float* __restrict__ AArow, const float* __restrict__ BBrow,
    const int* __restrict__ powerp, float* __restrict__ out, int N) {
  __shared__ float r0[TPB], r1[TPB], r2[TPB];
  const int tid = threadIdx.x;

  // phase 1: mean / std of y_pred_class (population std, ddof=0)
  float s1 = 0.0f, s2 = 0.0f;
  for (int k = tid; k < N; k += TPB) {
    const float x = y_pred_class[k];
    s1 += x; s2 += x * x;
  }
  r0[tid] = s1; r1[tid] = s2;
  __syncthreads();
  for (int s = TPB / 2; s > 0; s >>= 1) {
    if (tid < s) { r0[tid] += r0[tid + s]; r1[tid] += r1[tid + s]; }
    __syncthreads();
  }
  const float m  = r0[0] / (float)N;
  const float vr = r1[0] / (float)N - m * m;
  const float sd = sqrtf(fmaxf(vr, 0.0f));
  __syncthreads();

  // phase 2: focal sums
  float sOne = 0.0f, sCwf = 0.0f, sFb = 0.0f;
  for (int k = tid; k < N; k += TPB) {
    const float ypc = y_pred_class[k];
    const float yc  = y_class[k];
    const float t   = target[k];
    const float yp  = output[k];
    const float x   = fminf(fmaxf(yp, 1e-7f), 1.0f - 1e-7f);
    const float bce = -t * logf(x) - (1.0f - t) * logf(1.0f - x);
    const float nrm = fminf(fmaxf((ypc - m) / (2.0f * sd) + 0.5f, 0.0f), 1.0f);
    const float base = (1.0f - yc) * nrm;
    const float cwf  = base * base;  // GAMMA = 2
    sOne += (1.0f - yc);
    sCwf += cwf;
    sFb  += cwf * bce;
  }
  r0[tid] = sOne; r1[tid] = sCwf; r2[tid] = sFb;
  __syncthreads();
  for (int s = TPB / 2; s > 0; s >>= 1) {
    if (tid < s) { r0[tid] += r0[tid + s]; r1[tid] += r1[tid + s]; r2[tid] += r2[tid + s]; }
    __syncthreads();
  }
  const float focal_mean = (r0[0] / r1[0]) * r2[0] / (float)N;
  __syncthreads();

  // phase 3: mAB / mAA / mBB
  float sAB = 0.0f, sAA = 0.0f, sBB = 0.0f;
  for (int k = tid; k < N; k += TPB) {
    const float wk = w[k];
    sAB += ABrow[k] * wk;
    sAA += AArow[k] * wk;
    sBB += BBrow[k] * wk;
  }
  r0[tid] = sAB; r1[tid] = sAA; r2[tid] = sBB;
  __syncthreads();
  for (int s = TPB / 2; s > 0; s >>= 1) {
    if (tid < s) { r0[tid] += r0[tid + s]; r1[tid] += r1[tid + s]; r2[tid] += r2[tid + s]; }
    __syncthreads();
  }
  if (tid == 0) {
    const float mAB = r0[0] / (float)N;
    const float mAA = r1[0] / (float)N;
    const float mBB = r2[0] / (float)N;
    const int power = powerp[0];
    float d;
    if (power == 1) {
      d = mAB / sqrtf(fabsf(mAA * mBB) + 1e-12f);
    } else if (power == 2) {
      d = (mAB * mAB) / (fabsf(mAA * mBB) + 1e-12f);
    } else {
      d = powf(mAB / sqrtf(mAA * mBB) + 1e-12f, (float)power);
    }
    if (isnan(d) || d < 0.0f) d = 0.0f;
    out[0] = focal_mean + 1000.0f * d;
  }
}

extern "C" void kernel_launch(void* const* d_in, const int* in_sizes, int n_in,
                              void* d_out, int out_size, void* d_ws, size_t ws_size,
                              hipStream_t stream) {
  const int N = in_sizes[0];  // 4096
  const float* target = (const float*)d_in[0];
  const float* output = (const float*)d_in[1];
  const float* y_class = (const float*)d_in[2];
  const float* y_pred_class = (const float*)d_in[3];
  const float* v1 = (const float*)d_in[4];
  const float* v2 = (const float*)d_in[5];
  const float* w  = (const float*)d_in[6];
  const int*   power = (const int*)d_in[7];

  float* ws    = (float*)d_ws;
  float* aavg  = ws;
  float* bavg  = ws + (size_t)N;
  float* ABrow = ws + (size_t)2 * N;
  float* AArow = ws + (size_t)3 * N;
  float* BBrow = ws + (size_t)4 * N;
  float* scal  = ws + (size_t)5 * N;

  const int nRowBlocks = N / 16;
  row_avg_kernel<<<nRowBlocks, TPB, 0, stream>>>(v1, v2, w, aavg, bavg, N);
  mean_kernel<<<1, TPB, 0, stream>>>(aavg, bavg, w, scal, N);
  row_disco_kernel<<<nRowBlocks, TPB, 0, stream>>>(v1, v2, w, aavg, bavg, scal,
                                                   ABrow, AArow, BBrow, N);
  final_kernel<<<1, TPB, 0, stream>>>(target, output, y_class, y_pred_class, w,
                                      ABrow, AArow, BBrow, power,
                                      (float*)d_out, N);
}